// KNNLoss_74663711473732
// MI455X (gfx1250) — compile-verified
//
#include <hip/hip_runtime.h>
#include <hip/hip_bf16.h>
#include <math.h>

typedef __attribute__((ext_vector_type(2))) float v2f;
typedef __attribute__((ext_vector_type(8))) float v8f;

#define NPTS 8192
#define DIMS 128
#define KNN  8
#define WAVES_PER_BLK 8
#define CHUNK (NPTS / WAVES_PER_BLK)   // 1024 columns (64 tiles) per wave

// ---------------------------------------------------------------------------
// Kernel 1: row squared norms  sq[i] = ||x_i||^2
// ---------------------------------------------------------------------------
__global__ __launch_bounds__(256) void knn_sq_kernel(const float* __restrict__ x,
                                                     float* __restrict__ sq) {
    int i = blockIdx.x * blockDim.x + threadIdx.x;   // 8192 threads
    const float4* row = (const float4*)(x + (size_t)i * DIMS);
    float acc = 0.0f;
#pragma unroll
    for (int c = 0; c < DIMS / 4; ++c) {
        float4 v = row[c];
        acc += v.x * v.x + v.y * v.y + v.z * v.z + v.w * v.w;
    }
    sq[i] = acc;
}

// ---------------------------------------------------------------------------
// Kernel 2: one block (8 waves) per 16-row panel; each wave sweeps a 1024-col
// chunk with fp32 WMMA + fused sqrt/exp/top-8 epilogue; block-level combine.
// ---------------------------------------------------------------------------
__global__ __launch_bounds__(256) void knn_main_kernel(const float* __restrict__ x,
                                                       const float* __restrict__ sq,
                                                       float* __restrict__ loss) {
    __shared__ float lds[WAVES_PER_BLK][2048];   // 8 KB per wave (64 KB total)

    const int lane = threadIdx.x & 31;
    const int wave = threadIdx.x >> 5;
    const int m0   = blockIdx.x * 16;        // 16 rows owned by this block
    const int lm   = lane & 15;
    const int hi   = lane >> 4;              // half-wave id
    const int hi2  = hi * 2;
    float* wbuf = lds[wave];
    float* flat = &lds[0][0];

    const float FINF = __builtin_inff();

    // ---- A panel: 16 rows x 128 K, fp32 WMMA 16x16x4 A layout, kept in VGPRs
    v2f a[32];
    {
        const float* xA = x + (size_t)(m0 + lm) * DIMS + hi2;
#pragma unroll
        for (int c = 0; c < 32; ++c)
            a[c] = *(const v2f*)(xA + 4 * c);
    }

    // ---- per-lane state: sq of owned rows (C-layout rows), denom acc, top-8 d2
    float sqi[8];
#pragma unroll
    for (int v = 0; v < 8; ++v) sqi[v] = sq[m0 + v + 8 * hi];

    float denom[8];
    float t[8][8];                  // sorted ascending, 8 smallest d2 per (lane,row)
#pragma unroll
    for (int v = 0; v < 8; ++v) {
        denom[v] = 0.0f;
#pragma unroll
        for (int s8 = 0; s8 < 8; ++s8) t[v][s8] = FINF;
    }

    // ---- sweep this wave's 64 column tiles
    const int nbeg = wave * CHUNK;
    const int nend = nbeg + CHUNK;
    for (int n0 = nbeg; n0 < nend; n0 += 16) {
        const float* xB = x + (size_t)(n0 + lm) * DIMS + hi2;
        if (n0 + 16 < nend)
            __builtin_prefetch(x + (size_t)(n0 + 16 + lm) * DIMS, 0, 0);
        float sqj = sq[n0 + lm];

        v8f acc = {0.f, 0.f, 0.f, 0.f, 0.f, 0.f, 0.f, 0.f};
#pragma unroll
        for (int g = 0; g < 4; ++g) {
            v2f bb[8];
#pragma unroll
            for (int c = 0; c < 8; ++c)
                bb[c] = *(const v2f*)(xB + 4 * (g * 8 + c));
#pragma unroll
            for (int c = 0; c < 8; ++c)
                acc = __builtin_amdgcn_wmma_f32_16x16x4_f32(
                    false, a[g * 8 + c], false, bb[c], (short)0, acc, false, false);
        }

        // epilogue: d2 -> d -> exp(-d); exclude diagonal; top-8 ladder on d2
#pragma unroll
        for (int v = 0; v < 8; ++v) {
            float dot = acc[v];
            float d2  = sqi[v] + sqj - 2.0f * dot;
            d2 = fmaxf(d2, 0.0f);
            float dd = sqrtf(d2);
            float s  = __expf(-dd);
            bool diag = (n0 + lm) == (m0 + v + 8 * hi);
            denom[v] += diag ? 0.0f : s;
            float cand = diag ? FINF : d2;
#pragma unroll
            for (int s8 = 0; s8 < 8; ++s8) {
                float lo = fminf(t[v][s8], cand);
                cand     = fmaxf(t[v][s8], cand);
                t[v][s8] = lo;
            }
        }
    }

    // ================= intra-wave reduction (per-wave LDS region) ===========
    // phase 1: denominators (row-major 16x16)
#pragma unroll
    for (int v = 0; v < 8; ++v)
        wbuf[(v + 8 * hi) * 16 + lm] = denom[v];
    __syncthreads();

    float denom_r = 0.0f;
    if (lane < 16) {
#pragma unroll
        for (int c = 0; c < 16; ++c) denom_r += wbuf[lane * 16 + c];
    }
    __syncthreads();

    // phase 2: top-8 candidate dump (row-major 16 x 128), reuse the buffer
#pragma unroll
    for (int v = 0; v < 8; ++v) {
        int row = v + 8 * hi;
#pragma unroll
        for (int s8 = 0; s8 < 8; ++s8)
            wbuf[row * 128 + lm * 8 + s8] = t[v][s8];
    }
    __syncthreads();

    float tt[8];
#pragma unroll
    for (int s8 = 0; s8 < 8; ++s8) tt[s8] = FINF;
    if (lane < 16) {
        const float* rowc = wbuf + lane * 128;
        for (int c = 0; c < 128; ++c) {
            float cand = rowc[c];
#pragma unroll
            for (int s8 = 0; s8 < 8; ++s8) {
                float lo = fminf(tt[s8], cand);
                cand     = fmaxf(tt[s8], cand);
                tt[s8]   = lo;
            }
        }
    }
    __syncthreads();

    // ================= block-level combine across the 8 waves ===============
    // dtop[row][wave][slot] at flat[row*64 + wave*8 + slot]   (1024 floats)
    // dsum[wave][row]       at flat[1024 + wave*16 + row]     ( 128 floats)
    if (lane < 16) {
#pragma unroll
        for (int s8 = 0; s8 < 8; ++s8)
            flat[lane * 64 + wave * 8 + s8] = tt[s8];
        flat[1024 + wave * 16 + lane] = denom_r;
    }
    __syncthreads();

    if (wave == 0 && lane < 16) {
        int r = lane;
        float denom_total = 0.0f;
#pragma unroll
        for (int w = 0; w < WAVES_PER_BLK; ++w)
            denom_total += flat[1024 + w * 16 + r];

        float tf[8];
#pragma unroll
        for (int s8 = 0; s8 < 8; ++s8) tf[s8] = FINF;
        const float* cands = flat + r * 64;
        for (int c = 0; c < 64; ++c) {
            float cand = cands[c];
#pragma unroll
            for (int s8 = 0; s8 < 8; ++s8) {
                float lo = fminf(tf[s8], cand);
                cand     = fmaxf(tf[s8], cand);
                tf[s8]   = lo;
            }
        }
        float s8sum = 0.0f;
#pragma unroll
        for (int s8 = 0; s8 < 8; ++s8) s8sum += sqrtf(tf[s8]);
        loss[m0 + r] = s8sum * (1.0f / (float)KNN) + __logf(denom_total);
    }
}

// ---------------------------------------------------------------------------
// Kernel 3: deterministic fixed-order reduction of 8192 per-row losses
// ---------------------------------------------------------------------------
__global__ __launch_bounds__(256) void knn_reduce_kernel(const float* __restrict__ loss,
                                                         float* __restrict__ out) {
    __shared__ float buf[256];
    int tid = threadIdx.x;
    float acc = 0.0f;
    for (int i = tid; i < NPTS; i += 256) acc += loss[i];
    buf[tid] = acc;
    __syncthreads();
    for (int s = 128; s > 0; s >>= 1) {
        if (tid < s) buf[tid] += buf[tid + s];
        __syncthreads();
    }
    if (tid == 0) out[0] = buf[0] / (float)NPTS;
}

// ---------------------------------------------------------------------------
extern "C" void kernel_launch(void* const* d_in, const int* in_sizes, int n_in,
                              void* d_out, int out_size, void* d_ws, size_t ws_size,
                              hipStream_t stream) {
    const float* x = (const float*)d_in[0];
    float* sq   = (float*)d_ws;           // [8192]
    float* loss = sq + NPTS;              // [8192]
    float* out  = (float*)d_out;          // scalar

    knn_sq_kernel<<<NPTS / 256, 256, 0, stream>>>(x, sq);
    knn_main_kernel<<<NPTS / 16, 256, 0, stream>>>(x, sq, loss);
    knn_reduce_kernel<<<1, 256, 0, stream>>>(loss, out);
}